// SocialRecommender_87866440942279
// MI455X (gfx1250) — compile-verified
//
#include <hip/hip_runtime.h>

#define D 64
#define LRELU_SLOPE 0.01f
#define LN_EPS 1e-5f

typedef __attribute__((ext_vector_type(2))) float v2f;
typedef __attribute__((ext_vector_type(8))) float v8f;

// ---------------------------------------------------------------------------
// utility kernels
// ---------------------------------------------------------------------------
__global__ void fill_zero_f32(float* __restrict__ p, int n4) {
  int i = blockIdx.x * blockDim.x + threadIdx.x;
  if (i < n4) reinterpret_cast<float4*>(p)[i] = make_float4(0.f, 0.f, 0.f, 0.f);
}

__global__ void copy_f32(const float* __restrict__ src, float* __restrict__ dst, int n4) {
  int i = blockIdx.x * blockDim.x + threadIdx.x;
  if (i < n4) reinterpret_cast<float4*>(dst)[i] = reinterpret_cast<const float4*>(src)[i];
}

// ---------------------------------------------------------------------------
// COO SpMM: out[rows[e]] += vals[e] * x[cols[e]]   (rows of width D=64)
// 16 lanes per edge, 4 floats (one float4 gather) per lane.
// unsafeAtomicAdd -> global_atomic_add_f32 on gfx1250.
// ---------------------------------------------------------------------------
__global__ void spmm_coo_atomic(const int* __restrict__ rows, const int* __restrict__ cols,
                                const float* __restrict__ vals, const float* __restrict__ x,
                                float* __restrict__ out, int nnz) {
  long t = (long)blockIdx.x * blockDim.x + threadIdx.x;
  int e = (int)(t >> 4);
  if (e >= nnz) return;
  int q = ((int)t & 15) << 2;  // float offset within the row, multiple of 4
  int r = rows[e];
  int c = cols[e];
  float v = vals[e];
  float4 xv = *reinterpret_cast<const float4*>(x + (long)c * D + q);
  float* o = out + (long)r * D + q;
  unsafeAtomicAdd(o + 0, v * xv.x);
  unsafeAtomicAdd(o + 1, v * xv.y);
  unsafeAtomicAdd(o + 2, v * xv.z);
  unsafeAtomicAdd(o + 3, v * xv.w);
}

// ---------------------------------------------------------------------------
// In-place fused tall-skinny GEMM:  X[n,:] = leaky_relu(X[n,:] @ W^T + b)
// X: nrows x 64 (nrows % 16 == 0), W: 64x64 row-major, b: 64.
// One wave32 per 16-row tile. Uses V_WMMA_F32_16X16X4_F32:
//   A (16x4 f32):  lane L (m=L&15): vgpr0 = X[m][k0+2*(L>>4)], vgpr1 = next k
//   B (4x16 f32):  lane L (n=L&15): vgpr0 = W^T[k0+2*(L>>4)][n] = W[n][kq]
//   C/D (16x16):   lane L (n=L&15): vgpr r holds row m = r + 8*(L>>4)
// 16 k-steps per 16x16 tile, 4 output tiles -> 64 wmma per wave.
// ---------------------------------------------------------------------------
__global__ __launch_bounds__(256) void gemm64_wmma_bias_lrelu(
    float* __restrict__ X, const float* __restrict__ W, const float* __restrict__ b,
    int ntiles) {
  __shared__ float lds_w[D * D];        // 16 KB
  __shared__ float lds_x[8][16 * D];    // 32 KB (one 16x64 tile per wave)

  const int lane = threadIdx.x & 31;
  const int wave = threadIdx.x >> 5;

  // cooperative stage of W (1024 float4 across 256 threads)
#pragma unroll
  for (int j = 0; j < 4; ++j) {
    int i = threadIdx.x + j * 256;
    reinterpret_cast<float4*>(lds_w)[i] = reinterpret_cast<const float4*>(W)[i];
  }

  const int tile = blockIdx.x * 8 + wave;
  const bool active = tile < ntiles;
  if (active) {
    // stage this wave's 16x64 X tile: 2 lanes per row, 8 float4 per lane
    const int row0 = tile * 16;
    const int r = lane >> 1;
    const int cb = (lane & 1) * 32;
    const float4* src = reinterpret_cast<const float4*>(X + (long)(row0 + r) * D + cb);
    float4* dst = reinterpret_cast<float4*>(&lds_x[wave][r * D + cb]);
#pragma unroll
    for (int j = 0; j < 8; ++j) dst[j] = src[j];
  }
  __syncthreads();
  if (!active) return;

  const int row0 = tile * 16;
  const int m = lane & 15;
  const int hi = lane >> 4;
  const float* xt = lds_x[wave];

#pragma unroll
  for (int ob = 0; ob < D; ob += 16) {
    const int n = ob + m;
    v8f acc = {};
#pragma unroll
    for (int k0 = 0; k0 < D; k0 += 4) {
      const int kq = k0 + 2 * hi;
      v2f a, bb;
      a.x = xt[m * D + kq];
      a.y = xt[m * D + kq + 1];
      bb.x = lds_w[n * D + kq];
      bb.y = lds_w[n * D + kq + 1];
      acc = __builtin_amdgcn_wmma_f32_16x16x4_f32(
          /*neg_a=*/false, a, /*neg_b=*/false, bb,
          /*c_mod=*/(short)0, acc, /*reuse_a=*/false, /*reuse_b=*/false);
    }
    const float bias = b[n];
#pragma unroll
    for (int rr = 0; rr < 8; ++rr) {
      const int mloc = rr + 8 * hi;
      float v = acc[rr] + bias;
      v = v > 0.f ? v : LRELU_SLOPE * v;
      X[(long)(row0 + mloc) * D + n] = v;
    }
  }
}

// ---------------------------------------------------------------------------
// out[row,:] = LN(base[row,:] + delta[row,:]) * gamma + beta
// One wave32 per row, 2 elements per lane, shfl_xor reduction (wave32).
// Safe when out == base (each lane reads its 2 elements before any write).
// ---------------------------------------------------------------------------
__global__ __launch_bounds__(256) void residual_layernorm64(
    const float* __restrict__ base, const float* __restrict__ delta,
    const float* __restrict__ gamma, const float* __restrict__ beta,
    float* __restrict__ out, int nrows) {
  const int lane = threadIdx.x & 31;
  const int wave = threadIdx.x >> 5;
  const int row = blockIdx.x * 8 + wave;
  if (row >= nrows) return;
  const float* bp = base + (long)row * D;
  const float* dp = delta + (long)row * D;
  float x0 = bp[lane] + dp[lane];
  float x1 = bp[lane + 32] + dp[lane + 32];
  float s = x0 + x1;
  float ss = x0 * x0 + x1 * x1;
#pragma unroll
  for (int off = 16; off; off >>= 1) {
    s += __shfl_xor(s, off, 32);
    ss += __shfl_xor(ss, off, 32);
  }
  const float mean = s * (1.0f / D);
  const float var = ss * (1.0f / D) - mean * mean;
  const float rs = rsqrtf(var + LN_EPS);
  float* op = out + (long)row * D;
  op[lane] = (x0 - mean) * rs * gamma[lane] + beta[lane];
  op[lane + 32] = (x1 - mean) * rs * gamma[lane + 32] + beta[lane + 32];
}

// ---------------------------------------------------------------------------
// orchestration
// ---------------------------------------------------------------------------
extern "C" void kernel_launch(void* const* d_in, const int* in_sizes, int n_in,
                              void* d_out, int out_size, void* d_ws, size_t ws_size,
                              hipStream_t stream) {
  const float* user_emb = (const float*)d_in[0];
  const float* item_emb = (const float*)d_in[1];
  const int* ui_rows = (const int*)d_in[2];
  const int* ui_cols = (const int*)d_in[3];
  const float* ui_vals = (const float*)d_in[4];
  const int* soc_rows = (const int*)d_in[5];
  const int* soc_cols = (const int*)d_in[6];
  const float* soc_vals = (const float*)d_in[7];
  const float* ui_W = (const float*)d_in[8];
  const float* ui_b = (const float*)d_in[9];
  const float* soc_W = (const float*)d_in[10];
  const float* soc_b = (const float*)d_in[11];
  const float* ln_ui_g = (const float*)d_in[12];
  const float* ln_ui_b = (const float*)d_in[13];
  const float* ln_soc_g = (const float*)d_in[14];
  const float* ln_soc_b = (const float*)d_in[15];

  const int U = in_sizes[0] / D;
  const int I = in_sizes[1] / D;
  const int nnz_ui = in_sizes[2];
  const int nnz_soc = in_sizes[5];
  const long UD = (long)U * D;
  const long ID = (long)I * D;

  float* out = (float*)d_out;
  float* ui0 = out;
  float* ui1 = out + UD;
  float* ui2 = out + 2 * UD;
  float* soc0 = out + 3 * UD;
  float* soc1 = out + 4 * UD;
  float* soc2 = out + 5 * UD;
  float* ci_out = out + 6 * UD;  // item embedding slot (final layer, also used as live ci)

  float* ua = (float*)d_ws;      // U x D scratch (shared with social path)
  float* ia = ua + UD;           // I x D scratch

  auto zero = [&](float* p, long n) {
    int n4 = (int)(n >> 2);
    fill_zero_f32<<<(n4 + 255) / 256, 256, 0, stream>>>(p, n4);
  };
  auto copy = [&](const float* s, float* d, long n) {
    int n4 = (int)(n >> 2);
    copy_f32<<<(n4 + 255) / 256, 256, 0, stream>>>(s, d, n4);
  };
  auto spmm = [&](const int* r, const int* c, const float* v, const float* x, float* o,
                  int nnz) {
    long threads = (long)nnz * 16;
    spmm_coo_atomic<<<(int)((threads + 255) / 256), 256, 0, stream>>>(r, c, v, x, o, nnz);
  };
  auto gemm = [&](float* X, const float* W, const float* b, int nrows) {
    int ntiles = nrows / 16;
    gemm64_wmma_bias_lrelu<<<(ntiles + 7) / 8, 256, 0, stream>>>(X, W, b, ntiles);
  };
  auto resln = [&](const float* base, const float* delta, const float* g, const float* b,
                   float* o, int nrows) {
    residual_layernorm64<<<(nrows + 7) / 8, 256, 0, stream>>>(base, delta, g, b, o, nrows);
  };

  // layer-0 embeddings straight to output
  copy(user_emb, ui0, UD);
  copy(user_emb, soc0, UD);

  // ----- user-item dual propagation, layer 0 -----
  zero(ua, UD);
  spmm(ui_rows, ui_cols, ui_vals, item_emb, ua, nnz_ui);   // user <- items
  gemm(ua, ui_W, ui_b, U);
  resln(user_emb, ua, ln_ui_g, ln_ui_b, ui1, U);           // cu1

  zero(ia, ID);
  spmm(ui_cols, ui_rows, ui_vals, user_emb, ia, nnz_ui);   // item <- users (transpose)
  gemm(ia, ui_W, ui_b, I);
  resln(item_emb, ia, ln_ui_g, ln_ui_b, ci_out, I);        // ci1 (lives in output tail slot)

  // ----- user-item dual propagation, layer 1 -----
  zero(ua, UD);
  spmm(ui_rows, ui_cols, ui_vals, ci_out, ua, nnz_ui);     // uses ci1
  gemm(ua, ui_W + D * D, ui_b + D, U);
  resln(ui1, ua, ln_ui_g + D, ln_ui_b + D, ui2, U);        // cu2

  zero(ia, ID);
  spmm(ui_cols, ui_rows, ui_vals, ui1, ia, nnz_ui);        // uses cu1
  gemm(ia, ui_W + D * D, ui_b + D, I);
  resln(ci_out, ia, ln_ui_g + D, ln_ui_b + D, ci_out, I);  // ci2 (in-place, safe)

  // ----- social propagation, layer 0 -----
  zero(ua, UD);
  spmm(soc_rows, soc_cols, soc_vals, user_emb, ua, nnz_soc);
  gemm(ua, soc_W, soc_b, U);
  resln(user_emb, ua, ln_soc_g, ln_soc_b, soc1, U);        // cs1

  // ----- social propagation, layer 1 -----
  zero(ua, UD);
  spmm(soc_rows, soc_cols, soc_vals, soc1, ua, nnz_soc);
  gemm(ua, soc_W + D * D, soc_b + D, U);
  resln(soc1, ua, ln_soc_g + D, ln_soc_b + D, soc2, U);    // cs2
}